// CNNLayer_21526376087889
// MI455X (gfx1250) — compile-verified
//
#include <hip/hip_runtime.h>
#include <stdint.h>

typedef __attribute__((ext_vector_type(2))) float v2f;
typedef __attribute__((ext_vector_type(8))) float v8f;
typedef uint32_t u32;
typedef __attribute__((ext_vector_type(4))) u32 v4u;
typedef __attribute__((ext_vector_type(8))) u32 v8u;

namespace {
constexpr int kL      = 10;
constexpr int kC      = 16;
constexpr int kW0     = 1 << kL;       // 1024 spatial positions after input conv
constexpr int kElems  = kW0 * kC;      // 16384 floats (constant per level)
constexpr int kWElems = 16384;         // weight staging buffer (64 KB)
constexpr int kThreads = 256;          // 8 wave32
constexpr int kWaves   = kThreads / 32;
constexpr int kInSiz   = 8192;         // per batch element
constexpr int kOutSiz  = 8192;
}

struct Params {
  const float* filters[10];
  const float* biases[10];
};

__device__ __forceinline__ v8f wmma4(v2f a, v2f b, v8f c) {
  // D = A(16x4 f32) * B(4x16 f32) + C(16x16 f32)
  return __builtin_amdgcn_wmma_f32_16x16x4_f32(
      /*neg_a=*/false, a, /*neg_b=*/false, b,
      /*c_mod=*/(short)0, c, /*reuse_a=*/false, /*reuse_b=*/false);
}

__device__ __forceinline__ float relu(float v) { return v > 0.f ? v : 0.f; }

// Stage a [rows, cols] f32 weight matrix from global into LDS via the
// Tensor Data Mover (2-D D#, 2 SGPR groups). Issued by wave 0 only (TDM
// ignores EXEC); completion via TENSORcnt, visibility via barrier.
__device__ void tdm_stage(const float* __restrict__ Wg, int rows, int cols,
                          float* __restrict__ wlds) {
  if (threadIdx.x < 32) {
    const u32 lds = (u32)(uintptr_t)wlds;                 // LDS byte address
    const unsigned long long ga = (unsigned long long)(uintptr_t)Wg;
    v4u g0;
    g0[0] = 1u;                                           // count=1 (valid D#)
    g0[1] = lds;                                          // lds_addr
    g0[2] = (u32)ga;                                      // global_addr[31:0]
    g0[3] = ((u32)(ga >> 32) & 0x01FFFFFFu) | (2u << 30); // addr[56:32] | type=2
    v8u g1;
    g1[0] = 2u << 16;                                     // data_size=4B, mask=0
    g1[1] = ((u32)cols & 0xFFFFu) << 16;                  // tensor_dim0[15:0]
    g1[2] = (((u32)cols >> 16) & 0xFFFFu)
          | (((u32)rows & 0xFFFFu) << 16);                // dim0[31:16] | dim1[15:0]
    g1[3] = (((u32)rows >> 16) & 0xFFFFu)
          | (((u32)cols & 0xFFFFu) << 16);                // dim1[31:16] | tile_dim0
    g1[4] = (u32)rows & 0xFFFFu;                          // tile_dim1 (tile_dim2=0)
    g1[5] = (u32)cols;                                    // tensor_dim0_stride[31:0]
    g1[6] = 0u;                                           // stride0 hi | stride1 lo
    g1[7] = 0u;
    asm volatile("tensor_load_to_lds %0, %1" :: "s"(g0), "s"(g1) : "memory");
    asm volatile("s_wait_tensorcnt 0x0" ::: "memory");
  }
  __syncthreads();
}

// One butterfly conv level as a GEMM:  Y[M,N] = relu(A[M,K] * W[K,N] + bias)
// mode 0: first half  — A columns contiguous, direct store
// mode 1: second half — A columns shuffled in,  store shuffled out
// Wlds != nullptr: B fragments served from TDM-staged LDS copy.
__device__ void conv_gemm(const float* __restrict__ Alds,
                          const float* __restrict__ Wg,
                          const float* __restrict__ Wlds,
                          const float* __restrict__ biasg,
                          float* __restrict__ Ylds,
                          int M, int K, int N, int mode) {
  const int lane = threadIdx.x & 31;
  const int wave = threadIdx.x >> 5;
  const int half = lane >> 4;
  const int l16  = lane & 15;
  const int ntn  = N >> 4;
  const int ntiles = (M >> 4) * ntn;   // always 64 for this network
  if (!Wlds) __builtin_prefetch(Wg, 0, 1);  // global_prefetch_b8 (L2-hot weights)

  for (int tile = wave; tile < ntiles; tile += kWaves) {
    const int tm = tile / ntn;
    const int tn = tile - tm * ntn;
    const int mrow = (tm << 4) + l16;
    const int ncol = (tn << 4) + l16;
    v8f acc = {0.f, 0.f, 0.f, 0.f, 0.f, 0.f, 0.f, 0.f};

    for (int k0 = 0; k0 < K; k0 += 4) {
      const int ka = k0 + (half << 1);             // even; (ka, ka+1) pair
      int acol = ka;
      if (mode == 1) {
        // k = j*(K/2) + ix*C + c  ->  source col = (2*ix + j)*C + c
        const int j  = (ka >= (K >> 1)) ? 1 : 0;
        const int r  = ka & ((K >> 1) - 1);
        const int ix = r >> 4;
        const int c  = r & 15;
        acol = (ix << 5) + (j << 4) + c;           // c even -> pair stays contiguous
      }
      const v2f a = *(const v2f*)(Alds + mrow * K + acol);   // ds_load_b64
      v2f b;
      if (Wlds) {
        b.x = Wlds[ka * N + ncol];
        b.y = Wlds[(ka + 1) * N + ncol];
      } else {
        b.x = Wg[ka * N + ncol];
        b.y = Wg[(ka + 1) * N + ncol];
      }
      acc = wmma4(a, b, acc);
    }

    const float bias = biasg[ncol];
#pragma unroll
    for (int r = 0; r < 8; ++r) {
      const int m = (tm << 4) + (half << 3) + r;
      const float v = relu(acc[r] + bias);
      if (mode == 0) {
        Ylds[m * N + ncol] = v;
      } else {
        // out shuffle: col (2*ix+j)*C+d of row s -> row 2s+j, col ix*C+d
        const int q  = ncol >> 4;
        const int d  = ncol & 15;
        const int j  = q & 1;
        const int ix = q >> 1;
        Ylds[m * N + j * (N >> 1) + (ix << 4) + d] = v;
      }
    }
  }
}

// Input interpolation: [1024,8](global) x [8,16] -> LDS [1024,16]
__device__ void input_gemm(const float* __restrict__ gin,
                           const float* __restrict__ Wg,
                           const float* __restrict__ biasg,
                           float* __restrict__ Ylds) {
  const int lane = threadIdx.x & 31;
  const int wave = threadIdx.x >> 5;
  const int half = lane >> 4;
  const int l16  = lane & 15;
  const float bias = biasg[l16];
  for (int tm = wave; tm < 64; tm += kWaves) {
    const int mrow = (tm << 4) + l16;
    v8f acc = {0.f, 0.f, 0.f, 0.f, 0.f, 0.f, 0.f, 0.f};
#pragma unroll
    for (int k0 = 0; k0 < 8; k0 += 4) {
      const int ka = k0 + (half << 1);
      const v2f a = *(const v2f*)(gin + mrow * 8 + ka);      // global_load_b64
      v2f b;
      b.x = Wg[ka * kC + l16];
      b.y = Wg[(ka + 1) * kC + l16];
      acc = wmma4(a, b, acc);
    }
#pragma unroll
    for (int r = 0; r < 8; ++r) {
      const int m = (tm << 4) + (half << 3) + r;
      Ylds[m * kC + l16] = relu(acc[r] + bias);
    }
  }
}

// Middle switch: per (k,x) 16x16 dense; X in as [x, k*16+c], out as [k, x*16+d]
__device__ void mid_stage(const float* __restrict__ Xlds,
                          const float* __restrict__ dense,
                          const float* __restrict__ mbias,
                          float* __restrict__ Ylds) {
  for (int idx = threadIdx.x; idx < 32 * 32 * 16; idx += kThreads) {
    const int k   = idx >> 9;
    const int rem = idx & 511;
    const int x   = rem >> 4;
    const int d   = rem & 15;
    const float* Dm = dense + (((k << 5) + x) << 8);         // (k*32+x)*256
    const float* xr = Xlds + (x << 9) + (k << 4);
    float acc = mbias[(((k << 5) + x) << 4) + d];
#pragma unroll
    for (int c = 0; c < 16; ++c) acc += xr[c] * Dm[(c << 4) + d];
    Ylds[(k << 9) + (x << 4) + d] = relu(acc);
  }
}

// Output projection: [1024,16](LDS) x [16,8] -> global, no bias/relu
__device__ void output_gemm(const float* __restrict__ Xlds,
                            const float* __restrict__ Wg,
                            float* __restrict__ gout) {
  const int lane = threadIdx.x & 31;
  const int wave = threadIdx.x >> 5;
  const int half = lane >> 4;
  const int l16  = lane & 15;
  for (int tm = wave; tm < 64; tm += kWaves) {
    const int mrow = (tm << 4) + l16;
    v8f acc = {0.f, 0.f, 0.f, 0.f, 0.f, 0.f, 0.f, 0.f};
#pragma unroll
    for (int k0 = 0; k0 < kC; k0 += 4) {
      const int ka = k0 + (half << 1);
      const v2f a = *(const v2f*)(Xlds + mrow * kC + ka);
      v2f b;
      b.x = (l16 < 8) ? Wg[ka * 8 + l16] : 0.f;        // pad N 8->16 with zeros
      b.y = (l16 < 8) ? Wg[(ka + 1) * 8 + l16] : 0.f;
      acc = wmma4(a, b, acc);
    }
    if (l16 < 8) {
#pragma unroll
      for (int r = 0; r < 8; ++r) {
        const int m = (tm << 4) + (half << 3) + r;
        gout[m * 8 + l16] = acc[r];
      }
    }
  }
}

__global__ __launch_bounds__(kThreads)
void butterfly_fused(const float* __restrict__ in_data,
                     const float* __restrict__ in_filter,
                     const float* __restrict__ in_bias,
                     Params p,
                     const float* __restrict__ mid_dense,
                     const float* __restrict__ mid_bias,
                     const float* __restrict__ out_filter,
                     float* __restrict__ out) {
  __shared__ float bufA[kElems];    // 64 KB activations (ping)
  __shared__ float bufB[kElems];    // 64 KB activations (pong)
  __shared__ float wbuf[kWElems];   // 64 KB TDM-staged weights (192 KB < 320 KB/WGP)
  const int b = blockIdx.x;

  input_gemm(in_data + (size_t)b * kInSiz, in_filter, in_bias, bufA);
  __syncthreads();

  const float* src = bufA;
  float* dst = bufB;

  // First half: spatial halves, channels double.  GEMM [M, K] x [K, N]
  for (int lvl = 1; lvl <= kL / 2; ++lvl) {
    const int M = kW0 >> lvl;        // 512 .. 32
    const int KN = kC << lvl;        // 32 .. 512
    const bool stage = (KN * KN) <= kWElems;   // lvl 1..3 fit in 64 KB
    if (stage) tdm_stage(p.filters[lvl - 1], KN, KN, wbuf);
    conv_gemm(src, p.filters[lvl - 1], stage ? wbuf : nullptr,
              p.biases[lvl - 1], dst, M, KN, KN, /*mode=*/0);
    __syncthreads();
    const float* t = src; src = dst; dst = (float*)t;
  }

  mid_stage(src, mid_dense, mid_bias, dst);
  __syncthreads();
  { const float* t = src; src = dst; dst = (float*)t; }

  // Second half: spatial doubles, channel blocks halve, shuffles fused.
  for (int lvl = kL / 2 + 1; lvl <= kL; ++lvl) {
    const int M = 1 << (lvl - 1);            // 32 .. 512
    const int KN = kC << (kL - lvl + 1);     // 512 .. 32
    const bool stage = (KN * KN) <= kWElems;   // lvl 8..10 fit in 64 KB
    if (stage) tdm_stage(p.filters[lvl - 1], KN, KN, wbuf);
    conv_gemm(src, p.filters[lvl - 1], stage ? wbuf : nullptr,
              p.biases[lvl - 1], dst, M, KN, KN, /*mode=*/1);
    __syncthreads();
    const float* t = src; src = dst; dst = (float*)t;
  }

  output_gemm(src, out_filter, out + (size_t)b * kOutSiz);
}

extern "C" void kernel_launch(void* const* d_in, const int* in_sizes, int n_in,
                              void* d_out, int out_size, void* d_ws, size_t ws_size,
                              hipStream_t stream) {
  (void)in_sizes; (void)n_in; (void)out_size; (void)d_ws; (void)ws_size;
  const float* in_data    = (const float*)d_in[0];
  const float* in_filter  = (const float*)d_in[1];
  const float* in_bias    = (const float*)d_in[2];
  Params p;
  for (int i = 0; i < 10; ++i) {
    p.filters[i] = (const float*)d_in[3 + i];
    p.biases[i]  = (const float*)d_in[13 + i];
  }
  const float* mid_dense  = (const float*)d_in[23];
  const float* mid_bias   = (const float*)d_in[24];
  const float* out_filter = (const float*)d_in[25];

  butterfly_fused<<<256, kThreads, 0, stream>>>(
      in_data, in_filter, in_bias, p, mid_dense, mid_bias, out_filter,
      (float*)d_out);
}